// CDFNet_34866544509653
// MI455X (gfx1250) — compile-verified
//
#include <hip/hip_runtime.h>
#include <math.h>

#define DD   32
#define NN   128
#define MHH  5
#define BB   2048

typedef float v2f __attribute__((ext_vector_type(2)));
typedef float v8f __attribute__((ext_vector_type(8)));

// ---- helpers ------------------------------------------------------------

// softplus(10x)/10, numerically stable (matches jax.nn.softplus(10x)/10).
// Precise libm — used only in the tiny one-shot prep/reduce kernels.
static __device__ __forceinline__ float sp10(float x) {
    float z = 10.0f * x;
    return (fmaxf(z, 0.0f) + log1pf(expf(-fabsf(z)))) * 0.1f;
}

// Fast branchless tanh: native V_TANH_F32 on gfx1250 if the builtin exists,
// else tanh(x) = 1 - 2/(exp(2x)+1) via native v_exp_f32 / v_rcp_f32.
static __device__ __forceinline__ float fast_tanh(float x) {
#if __has_builtin(__builtin_amdgcn_tanhf)
    return __builtin_amdgcn_tanhf(x);
#else
    float e = __expf(2.0f * x);
    return 1.0f - 2.0f * __builtin_amdgcn_rcpf(e + 1.0f);
#endif
}

// swap lane halves (lane ^ 16) via ds_swizzle group-of-32:
// offset = or<<10 | xor<<5 | and = (0<<10)|(0x10<<5)|0x1F = 0x21F
static __device__ __forceinline__ float swap16(float x) {
    int r = __builtin_amdgcn_ds_swizzle(__float_as_int(x), 0x021F);
    return __int_as_float(r);
}

static __device__ __forceinline__ v8f wmma_f32_4(v2f a, v2f b, v8f c) {
    return __builtin_amdgcn_wmma_f32_16x16x4_f32(
        /*neg_a=*/false, a, /*neg_b=*/false, b,
        /*c_mod=*/(short)0, c, /*reuse_a=*/false, /*reuse_b=*/false);
}

// ---- kernel 1: weight preprocessing ------------------------------------

__global__ __launch_bounds__(256) void cdf_prep(
    const float* __restrict__ w_first, const float* __restrict__ w_mid,
    const float* __restrict__ w_last,  const float* __restrict__ a_in,
    float* __restrict__ Wf, float* __restrict__ Wm,
    float* __restrict__ Wl, float* __restrict__ ta)
{
    int i = blockIdx.x * 256 + threadIdx.x;
    if (i < 4 * DD * NN * MHH * MHH) Wm[i] = sp10(w_mid[i]);     // 409600
    if (i < DD * NN * MHH) { Wf[i] = sp10(w_first[i]);            // 20480
                             Wl[i] = sp10(w_last[i]); }
    if (i < 5 * DD * NN * MHH) ta[i] = tanhf(a_in[i]);            // 102400
}

// ---- kernel 2: main network (WMMA f32 16x16x4) --------------------------
// One wave handles one (n, 16-row batch chunk); loops over d accumulating
// sum_d log(phidot) and sum_d -log(phidot+1e-10) in registers.
// State orientation: S[m]/P[m] hold phi^T / phidot^T, batch column = lane
// (lanes 0-15 valid; lanes 16-31 carry harmless finite garbage).

__global__ __launch_bounds__(256) void cdf_main(
    const float* __restrict__ X,
    const float* __restrict__ Wf, const float* __restrict__ Wm,
    const float* __restrict__ Wl, const float* __restrict__ ta,
    const float* __restrict__ b_in, const float* __restrict__ b_last,
    float* __restrict__ slog, float* __restrict__ sneg)
{
    const int lane = threadIdx.x & 31;
    int w = __builtin_amdgcn_readfirstlane((int)(blockIdx.x * 8u + (threadIdx.x >> 5)));
    const int n = w & (NN - 1);          // network index
    const int c = w >> 7;                // batch chunk (16 rows)
    const int l15 = lane & 15;
    const bool lo  = lane < 16;
    const int row  = c * 16 + l15;       // batch row for this lane

    float acc_slog = 0.0f, acc_sneg = 0.0f;

    for (int d = 0; d < DD; ++d) {
        const int dn = d * NN + n;

        // ---- first layer: k=1 -> 5 (pure VALU, scalar weights) ----
        float x = X[row * DD + d];
        float S[5], P[5];
        #pragma unroll
        for (int m = 0; m < 5; ++m) {
            float wf = Wf[dn * 5 + m];
            float b0 = b_in[dn * 5 + m];        // layer 0 bias
            float t0 = ta[dn * 5 + m];          // tanh(a) layer 0
            float pre  = fmaf(x, wf, b0);
            float th   = fast_tanh(pre);
            float gate = fmaf(t0, 1.0f - th * th, 1.0f);
            S[m] = fmaf(th, t0, pre);
            P[m] = wf * gate;
        }

        // ---- 4 middle layers: 5->5 via two chained WMMA f32 16x16x4 ----
        #pragma unroll
        for (int li = 0; li < 4; ++li) {
            const int wbase = ((li * DD + d) * NN + n) * 25;       // W[k][m]
            const int sbase = ((li + 1) * DD * NN + dn) * 5;       // bias/ta

            // A = W^T : A[m=lane][k], lanes16-31 carry k=2,3; 2nd WMMA k=4
            const bool mok = l15 < 5;
            v2f A, A2;
            A.x  = mok ? Wm[wbase + (lo ? 0 : 10) + l15] : 0.0f;   // k0*5
            A.y  = mok ? Wm[wbase + (lo ? 5 : 15) + l15] : 0.0f;   // k1*5
            A2.x = (lo && mok) ? Wm[wbase + 20 + l15] : 0.0f;      // k=4
            A2.y = 0.0f;

            // B = state^T: VGPR0 rows {0,2}, VGPR1 rows {1,3}; 2nd: row {4}
            float s2h = swap16(S[2]), s3h = swap16(S[3]);
            float p2h = swap16(P[2]), p3h = swap16(P[3]);
            v2f Bs, Bs2, Bp, Bp2;
            Bs.x  = lo ? S[0] : s2h;   Bs.y  = lo ? S[1] : s3h;
            Bs2.x = lo ? S[4] : 0.0f;  Bs2.y = 0.0f;
            Bp.x  = lo ? P[0] : p2h;   Bp.y  = lo ? P[1] : p3h;
            Bp2.x = lo ? P[4] : 0.0f;  Bp2.y = 0.0f;

            // C seeded with bias (phi path), zeros (phidot path)
            v8f cb = {};
            #pragma unroll
            for (int m = 0; m < 5; ++m) cb[m] = b_in[sbase + m];
            v8f cz = {};

            v8f pre = wmma_f32_4(A,  Bs,  cb);
            pre     = wmma_f32_4(A2, Bs2, pre);
            v8f pd  = wmma_f32_4(A,  Bp,  cz);
            pd      = wmma_f32_4(A2, Bp2, pd);

            #pragma unroll
            for (int m = 0; m < 5; ++m) {
                float t    = ta[sbase + m];
                float pr   = pre[m];
                float th   = fast_tanh(pr);
                float gate = fmaf(t, 1.0f - th * th, 1.0f);
                S[m] = fmaf(th, t, pr);
                P[m] = pd[m] * gate;
            }
        }

        // ---- last layer: 5 -> 1, sigmoid-derivative gate (VALU) ----
        float preL = b_last[dn];
        float pdL  = 0.0f;
        #pragma unroll
        for (int m = 0; m < 5; ++m) {
            float wl = Wl[dn * 5 + m];
            preL = fmaf(S[m], wl, preL);
            pdL  = fmaf(P[m], wl, pdL);
        }
        // sigmoid via native exp + rcp (branchless)
        float e  = __expf(-preL);
        float sg = __builtin_amdgcn_rcpf(1.0f + e);
        float phid = pdL * sg * (1.0f - sg);     // strictly > 0
        acc_slog += __logf(phid);
        acc_sneg -= __logf(phid + 1e-10f);
    }

    if (lo) {
        slog[row * NN + n] = acc_slog;
        sneg[row * NN + n] = acc_sneg;
    }
}

// ---- kernel 3: per-batch-row reduction ---------------------------------

__global__ __launch_bounds__(128) void cdf_reduce(
    const float* __restrict__ slog, const float* __restrict__ sneg,
    const float* __restrict__ a_last, float* __restrict__ out)
{
    __shared__ float sm[128];
    __shared__ float sm2[128];
    const int b  = blockIdx.x;
    const int nt = threadIdx.x;   // n index

    // fm = min_n mean_d ( -log(phidot + 1e-10) )
    sm[nt] = sneg[b * NN + nt] * (1.0f / 32.0f);
    __syncthreads();
    for (int s = 64; s > 0; s >>= 1) {
        if (nt < s) sm[nt] = fminf(sm[nt], sm[nt + s]);
        __syncthreads();
    }
    float fm = sm[0];
    __syncthreads();

    // prod_n = prod_d exp(fm)*phidot = exp(32*fm + sum_d log(phidot))
    float prod = expf(fm * 32.0f + slog[b * NN + nt]);
    float al   = sp10(a_last[nt]);
    sm[nt]  = al;
    sm2[nt] = prod * al;
    __syncthreads();
    for (int s = 64; s > 0; s >>= 1) {
        if (nt < s) { sm[nt] += sm[nt + s]; sm2[nt] += sm2[nt + s]; }
        __syncthreads();
    }
    if (nt == 0) {
        float lk = sm2[0] / sm[0];
        out[b] = logf(lk + 1e-10f) - fm * 32.0f;
    }
}

// ---- launcher -----------------------------------------------------------

extern "C" void kernel_launch(void* const* d_in, const int* in_sizes, int n_in,
                              void* d_out, int out_size, void* d_ws, size_t ws_size,
                              hipStream_t stream) {
    (void)in_sizes; (void)n_in; (void)out_size; (void)ws_size;
    const float* X       = (const float*)d_in[0];
    const float* w_first = (const float*)d_in[1];
    const float* w_mid   = (const float*)d_in[2];
    const float* w_last  = (const float*)d_in[3];
    const float* b_in    = (const float*)d_in[4];
    const float* b_last  = (const float*)d_in[5];
    const float* a_in    = (const float*)d_in[6];
    const float* a_last  = (const float*)d_in[7];

    // workspace layout (floats): Wf | Wl | ta | Wm | slog | sneg  (~4.1 MB)
    float* ws   = (float*)d_ws;
    float* Wf   = ws;                 // 20480
    float* Wl   = ws + 20480;         // 20480
    float* ta   = ws + 40960;         // 102400
    float* Wm   = ws + 143360;        // 409600
    float* slog = ws + 552960;        // 262144
    float* sneg = ws + 815104;        // 262144

    cdf_prep<<<1600, 256, 0, stream>>>(w_first, w_mid, w_last, a_in, Wf, Wm, Wl, ta);
    cdf_main<<<2048, 256, 0, stream>>>(X, Wf, Wm, Wl, ta, b_in, b_last, slog, sneg);
    cdf_reduce<<<BB, 128, 0, stream>>>(slog, sneg, a_last, (float*)d_out);
}